// privacyLoss1_10763188043951
// MI455X (gfx1250) — compile-verified
//
#include <hip/hip_runtime.h>
#include <hip/hip_bf16.h>

// CDNA5 / gfx1250. Algebraically-reduced pairwise class loss:
//   result = par * [ (2*T_same_sq - T_all_sq) - 2*(2*T_same_G - T_all_G) ]
// Terms come from per-class column sums S[c,d], per-class column sums of
// squares Q[c,d], class counts, and two endpoint-row corrections.
// S and Q are one-hot GEMMs computed with V_WMMA_F32_16X16X4_F32:
// M=16 classes, N=16 columns, K=4 samples per instruction, exact fp32.
//
// k1 processes 64 columns per wave as 4 interleaved WMMA sub-tiles so that
// every lane's B operand comes from a single float4 (global_load_b128) and
// the one-hot A matrix is amortized over 8 WMMAs per 4-sample step.

typedef float v2f __attribute__((ext_vector_type(2)));
typedef float v4f __attribute__((ext_vector_type(4)));
typedef float v8f __attribute__((ext_vector_type(8)));

#ifndef PRIV_CHUNKS
#define PRIV_CHUNKS 64
#endif

// ---------------------------------------------------------------------------
// Kernel 1: per-(64-column group, sample-chunk) one-hot WMMA accumulation.
// Block = 32 threads (one wave32, EXEC all ones -> WMMA legal).
//   A (16x4 f32, one-hot labels): VGPR0 = K0 (lanes 0-15) / K2 (lanes 16-31),
//                                 VGPR1 = K1 / K3, M = lane%16 (class).
//   B (4x16 f32): same K striping; sub-tile j uses element j of each float4,
//                 i.e. actual column = colbase + 4*(lane%16) + j.
//   C/D (16x16 f32): VGPR r -> M = r (lanes 0-15) / r+8 (lanes 16-31).
// ---------------------------------------------------------------------------
__global__ void __launch_bounds__(32)
k1_classsum(const float* __restrict__ x, const int* __restrict__ label,
            float* __restrict__ PS, float* __restrict__ PQ,
            int D, int spc)
{
    const int grp   = blockIdx.x;          // 64-column group
    const int chunk = blockIdx.y;          // sample chunk (spc samples)
    const int lane  = threadIdx.x;
    const int half  = lane >> 4;           // 0: K0/K1 rows, 1: K2/K3 rows
    const int c15   = lane & 15;           // class (A) / N index (B,C,D)
    const int col4  = grp * 64 + 4 * c15;  // first of this lane's 4 columns
    const int base0 = chunk * spc;

    v8f accS[4] = {};   // class sums of x,   sub-tiles j = 0..3
    v8f accQ[4] = {};   // class sums of x^2, sub-tiles j = 0..3

    for (int it = 0; it < spc; it += 4) {
        const int r0 = base0 + it + 2 * half;   // sample for K = 0 or 2
        const int r1 = r0 + 1;                  // sample for K = 1 or 3

        const v4f bx4 = *(const v4f*)(x + (size_t)r0 * D + col4);
        const v4f by4 = *(const v4f*)(x + (size_t)r1 * D + col4);
        const int l0  = label[r0];
        const int l1  = label[r1];

        v2f a;
        a.x = (l0 == c15) ? 1.0f : 0.0f;
        a.y = (l1 == c15) ? 1.0f : 0.0f;

        #pragma unroll
        for (int j = 0; j < 4; ++j) {
            v2f b, bq;
            b.x  = bx4[j];          b.y  = by4[j];
            bq.x = bx4[j] * bx4[j]; bq.y = by4[j] * by4[j];
            accS[j] = __builtin_amdgcn_wmma_f32_16x16x4_f32(
                          false, a, false, b,  (short)0, accS[j], false, false);
            accQ[j] = __builtin_amdgcn_wmma_f32_16x16x4_f32(
                          false, a, false, bq, (short)0, accQ[j], false, false);
        }
    }

    // Store: sub-tile j, lane c15, VGPR r -> S[row][col4 + j] (row-major),
    // so each lane writes one float4 per row per matrix.
    #pragma unroll
    for (int r = 0; r < 8; ++r) {
        const int row = r + 8 * half;                 // class index
        const size_t idx = ((size_t)chunk * 16 + row) * (size_t)D + col4;
        v4f vs, vq;
        #pragma unroll
        for (int j = 0; j < 4; ++j) { vs[j] = accS[j][r]; vq[j] = accQ[j][r]; }
        *(v4f*)(PS + idx) = vs;
        *(v4f*)(PQ + idx) = vq;
    }
}

// ---------------------------------------------------------------------------
// Kernel 2: deterministic reduction of chunk partials -> final S and Q.
// part layout: [matrix m (0=S,1=Q)][chunk][16][D]; fin: [m][16][D].
// ---------------------------------------------------------------------------
__global__ void k2_reduce(const float* __restrict__ part,
                          float* __restrict__ fin,
                          int perMat, int chunks)
{
    const int e = blockIdx.x * blockDim.x + threadIdx.x;
    if (e >= 2 * perMat) return;
    const int m  = e / perMat;
    const int rd = e - m * perMat;
    const float* p = part + (size_t)m * chunks * perMat + rd;
    float s = 0.0f;
    for (int ch = 0; ch < chunks; ++ch)
        s += p[(size_t)ch * perMat];
    fin[e] = s;
}

// ---------------------------------------------------------------------------
// Kernel 3: single-block final combine (fixed-tree reductions, deterministic).
// ---------------------------------------------------------------------------
__device__ __forceinline__ float blockReduce256(float v, float* s, int tid)
{
    s[tid] = v;
    __syncthreads();
    #pragma unroll
    for (int off = 128; off > 0; off >>= 1) {
        if (tid < off) s[tid] += s[tid + off];
        __syncthreads();
    }
    const float r = s[0];
    __syncthreads();
    return r;
}

__global__ void __launch_bounds__(256)
k3_final(const float* __restrict__ x, const int* __restrict__ label,
         const float* __restrict__ S, const float* __restrict__ Q,
         float* __restrict__ outp, int N, int D)
{
    __shared__ float red[256];
    __shared__ int   cnt[16];
    __shared__ float qsum[10];
    const int tid = threadIdx.x;

    if (tid < 16) cnt[tid] = 0;
    __syncthreads();
    for (int i = tid; i < N; i += 256)
        atomicAdd(&cnt[label[i] & 15], 1);     // integer: order-independent
    __syncthreads();

    const int l0 = label[0] & 15;
    const int lN = label[N - 1] & 15;

    float tg = 0.0f;   // T_same_G  = sum_d sum_c SA_c(d)*SB_c(d)
    float ta = 0.0f;   // T_all_G   = sum_d (rowsum - xN)*(rowsum - x0)
    float s0 = 0.0f;   // ||x_0||^2
    float sN = 0.0f;   // ||x_{N-1}||^2

    for (int d = tid; d < D; d += 256) {
        const float x0 = x[d];
        const float xN = x[(size_t)(N - 1) * D + d];
        s0 += x0 * x0;
        sN += xN * xN;
        float rowsum = 0.0f, tgd = 0.0f;
        #pragma unroll
        for (int c = 0; c < 10; ++c) {
            const float Sc = S[(size_t)c * D + d];
            const float SA = Sc - ((c == lN) ? xN : 0.0f);  // rows 0..N-2
            const float SB = Sc - ((c == l0) ? x0 : 0.0f);  // cols 1..N-1
            tgd    += SA * SB;
            rowsum += Sc;
        }
        tg += tgd;
        ta += (rowsum - xN) * (rowsum - x0);
    }
    tg = blockReduce256(tg, red, tid);
    ta = blockReduce256(ta, red, tid);
    s0 = blockReduce256(s0, red, tid);
    sN = blockReduce256(sN, red, tid);

    for (int c = 0; c < 10; ++c) {
        float q = 0.0f;
        for (int d = tid; d < D; d += 256)
            q += Q[(size_t)c * D + d];
        q = blockReduce256(q, red, tid);
        if (tid == 0) qsum[c] = q;
    }
    __syncthreads();

    if (tid == 0) {
        double T_same_sq = 0.0, SqAtot = 0.0, SqBtot = 0.0;
        for (int c = 0; c < 10; ++c) {
            const double nA  = (double)cnt[c] - ((c == lN) ? 1.0 : 0.0);
            const double nB  = (double)cnt[c] - ((c == l0) ? 1.0 : 0.0);
            const double SqA = (double)qsum[c] - ((c == lN) ? (double)sN : 0.0);
            const double SqB = (double)qsum[c] - ((c == l0) ? (double)s0 : 0.0);
            T_same_sq += nB * SqA + nA * SqB;
            SqAtot += SqA;
            SqBtot += SqB;
        }
        const double T_all_sq = (double)(N - 1) * (SqAtot + SqBtot);
        const double gterm    = 2.0 * (2.0 * (double)tg - (double)ta);
        const double total    = (2.0 * T_same_sq - T_all_sq) - gterm;
        const double par      = 0.5 / (double)N;   // SIGMA*0.5 / N
        outp[0] = (float)(par * total);
    }
}

// ---------------------------------------------------------------------------
extern "C" void kernel_launch(void* const* d_in, const int* in_sizes, int n_in,
                              void* d_out, int out_size, void* d_ws, size_t ws_size,
                              hipStream_t stream)
{
    const float* x     = (const float*)d_in[0];
    const int*   label = (const int*)d_in[1];

    const int N = in_sizes[1];              // 4096
    const int D = in_sizes[0] / N;          // 1024
    const int groups = D / 64;              // 16  (64 columns per wave)
    const int chunks = PRIV_CHUNKS;         // 64
    const int spc    = N / chunks;          // 64 samples per chunk

    const int perMat = 16 * D;              // floats per S or Q matrix
    float* ws  = (float*)d_ws;
    float* PS  = ws;                                        // [chunks][16][D]
    float* PQ  = ws + (size_t)chunks * perMat;              // [chunks][16][D]
    float* FIN = ws + 2 * (size_t)chunks * perMat;          // S then Q finals

    dim3 g1(groups, chunks);
    k1_classsum<<<g1, 32, 0, stream>>>(x, label, PS, PQ, D, spc);

    const int tot = 2 * perMat;
    k2_reduce<<<(tot + 255) / 256, 256, 0, stream>>>(ws, FIN, perMat, chunks);

    k3_final<<<1, 256, 0, stream>>>(x, label, FIN, FIN + perMat,
                                    (float*)d_out, N, D);
}